// PrototypeContrastLoss_54417235640829
// MI455X (gfx1250) — compile-verified
//
#include <hip/hip_runtime.h>
#include <math.h>

// PrototypeContrastLoss for MI455X (gfx1250, wave32).
//
// Roofline: feat is 210 MB fp32 -> ~9.7us at 23.3 TB/s HBM; the prototype
// GEMM is only ~3.4 GFLOP -> memory bound. Single pass over feat: DMA a
// 128px x 256ch tile into LDS with the Tensor Data Mover (padded rows give a
// conflict-free 132-float stride), normalize in-LDS, then accumulate
// k0[19,256] with V_WMMA_F32_16X16X4_F32 (full f32 precision; pos is 0/1).

#define TAUF 0.07f
#define EPSF 1e-12f

constexpr int Bn   = 8;
constexpr int Cc   = 256;
constexpr int Kc   = 19;
constexpr int HW   = 160 * 160;        // 25600
constexpr int TILE = 128;              // pixels per block (HW % TILE == 0)
constexpr int LSTR = 132;              // LDS row stride in floats: 132*4=528 (16B aligned),
                                       // 132%64==4 -> conflict-free column reads
constexpr int NBLK = (Bn * HW) / TILE; // 1600 blocks

typedef __attribute__((ext_vector_type(2))) float    v2f;
typedef __attribute__((ext_vector_type(8))) float    v8f;
typedef __attribute__((ext_vector_type(4))) unsigned v4u;
typedef __attribute__((ext_vector_type(8))) int      v8i;
typedef __attribute__((ext_vector_type(4))) int      v4i;

// ---------------------------------------------------------------- init
__global__ void init_ws_kernel(float* __restrict__ proto, unsigned* __restrict__ amask) {
    int i = blockIdx.x * blockDim.x + threadIdx.x;
    if (i < 32 * Cc) proto[i] = 0.0f;
    if (i == 0) *amask = 0u;
}

// ---------------------------------------------------------------- main
__global__ __launch_bounds__(256)
void proto_accum_kernel(const float* __restrict__ feat, const int* __restrict__ gt,
                        float* __restrict__ proto, unsigned* __restrict__ amask) {
    __shared__ float    sF[Cc * LSTR];   // 135168 B: feat tile, [channel][pixel], padded rows
    __shared__ float    sR[TILE];        // per-pixel 1/max(||f||, eps)
    __shared__ unsigned sM[TILE];        // per-pixel 19-bit class mask
    __shared__ unsigned sOr;

    const int tid = threadIdx.x;
    const int blk = blockIdx.x;
    const int b   = blk / (HW / TILE);
    const int hw0 = (blk % (HW / TILE)) * TILE;

    if (tid == 0) sOr = 0u;

    const float* fbase = feat + (size_t)b * Cc * HW + hw0;

#if __has_builtin(__builtin_amdgcn_tensor_load_to_lds)
    // ---- TDM: one descriptor DMAs the whole 256-row x 128-float tile.
    // D# per CDNA5 ISA. Pad fields reproduce the
    // LSTR=132 layout: pad_interval=6 (every 128 DWORDs) + pad_amount=3 (4 DWORDs).
    if (tid < 32) {                       // wave 0 issues; EXEC ignored by TDM
        unsigned long long ga = (unsigned long long)(uintptr_t)fbase;
        unsigned lds_off      = (unsigned)(uintptr_t)&sF[0];   // low 32 bits = LDS byte offset

        v4u g0;
        g0[0] = 1u;                                   // count=1, user mode, no gather
        g0[1] = lds_off;                              // lds_addr
        g0[2] = (unsigned)(ga & 0xFFFFFFFFu);         // global_addr[31:0]
        g0[3] = (unsigned)((ga >> 32) & 0x01FFFFFFu)  // global_addr[56:32]
              | (2u << 30);                           // type=2 ("image")

        v8i g1;
        g1[0] = (int)((2u << 16)        // data_size = 4 bytes
              |       (1u << 20)        // pad_enable
              |       (6u << 22)        // pad_interval: 128 DWORDs
              |       (3u << 25));      // pad_amount : 4 DWORDs (16 B)
        g1[1] = (int)((unsigned)(HW & 0xFFFF) << 16);       // tensor_dim0[15:0] (25600)
        g1[2] = (int)(((unsigned)HW >> 16) | ((unsigned)Cc << 16)); // dim0 hi | tensor_dim1 lo
        g1[3] = (int)((unsigned)TILE << 16);                // tensor_dim1 hi=0 | tile_dim0=128
        g1[4] = (int)(unsigned)Cc;                          // tile_dim1=256 | tile_dim2=0
        g1[5] = (int)(unsigned)HW;                          // tensor_dim0_stride[31:0]
        g1[6] = 0;                                          // stride hi | dim1_stride lo
        g1[7] = 0;

        v4i gz4 = {0, 0, 0, 0};                 // 2-D tile: groups 2/3 unused
        v8i gz8 = {0, 0, 0, 0, 0, 0, 0, 0};     // trailing descriptor group unused
        // amdgpu-toolchain (clang-23) 6-arg form
        __builtin_amdgcn_tensor_load_to_lds(g0, g1, gz4, gz4, gz8, 0);
    }
#else
    // ---- fallback: coalesced float4 staging through VGPRs
    for (int it = 0; it < (Cc * TILE) / (256 * 4); ++it) {
        int e  = (it * 256 + tid) * 4;
        int ch = e >> 7;
        int px = e & (TILE - 1);
        float4 v = *(const float4*)(fbase + (size_t)ch * HW + px);
        *(float4*)(&sF[ch * LSTR + px]) = v;
    }
#endif

    // ---- overlap with the DMA: class masks from gt (independent of sF)
    {
        int p = tid & (TILE - 1);
        if (tid < TILE) {
            unsigned m = 0u;
            const int* gbase = gt + (size_t)b * Kc * HW + hw0 + p;
            #pragma unroll
            for (int k = 0; k < Kc; ++k)
                m |= (gbase[(size_t)k * HW] == 1) ? (1u << k) : 0u;
            sM[p] = m;
            atomicOr(&sOr, m);
        }
    }

#if __has_builtin(__builtin_amdgcn_tensor_load_to_lds)
    if (tid < 32) __builtin_amdgcn_s_wait_tensorcnt(0);
#endif
    __syncthreads();

    // ---- per-pixel reciprocal norm: 2 threads split a pixel's 256 channels
    {
        int p    = tid >> 1;
        int cbeg = (tid & 1) * (Cc / 2);
        float s  = 0.0f;
        for (int c = 0; c < Cc / 2; ++c) {
            float v = sF[(cbeg + c) * LSTR + p];
            s += v * v;
        }
        s += __shfl_xor(s, 1);
        if ((tid & 1) == 0) sR[p] = 1.0f / fmaxf(sqrtf(s), EPSF);
    }
    __syncthreads();
    if (tid == 0) atomicOr(amask, sOr);

    // ---- WMMA: K0[32x256] += pos^T(32 x TILE) @ featn(TILE x 256)
    // A 16x4 f32: lanes 0-15 rows M, vgpr0=K0/vgpr1=K1; lanes 16-31: K2/K3.
    // B 4x16 f32: lanes 0-15 cols N, vgpr0=K0/vgpr1=K1; lanes 16-31: K2/K3.
    const int lane = tid & 31;
    const int wid  = tid >> 5;            // 8 waves, each owns 32 channels
    const int ln   = lane & 15;
    const int kb   = (lane >> 4) * 2;     // 0 or 2: which K pair this half-wave packs
    const int ch0  = wid * 32 + ln;
    const int ch1  = ch0 + 16;

    v8f acc00 = {}, acc01 = {}, acc10 = {}, acc11 = {};
    for (int s4 = 0; s4 < TILE; s4 += 4) {
        int px = s4 + kb;
        unsigned m0 = sM[px], m1 = sM[px + 1];
        float    r0 = sR[px], r1 = sR[px + 1];
        v2f a0, a1, b0, b1;
        a0.x = (float)((m0 >> ln) & 1u);          // classes 0..15
        a0.y = (float)((m1 >> ln) & 1u);
        a1.x = (float)((m0 >> (ln + 16)) & 1u);   // classes 16..18 (bits>=19 are 0)
        a1.y = (float)((m1 >> (ln + 16)) & 1u);
        b0.x = sF[ch0 * LSTR + px    ] * r0;
        b0.y = sF[ch0 * LSTR + px + 1] * r1;
        b1.x = sF[ch1 * LSTR + px    ] * r0;
        b1.y = sF[ch1 * LSTR + px + 1] * r1;
        acc00 = __builtin_amdgcn_wmma_f32_16x16x4_f32(false, a0, false, b0, (short)0, acc00, false, false);
        acc01 = __builtin_amdgcn_wmma_f32_16x16x4_f32(false, a0, false, b1, (short)0, acc01, false, false);
        acc10 = __builtin_amdgcn_wmma_f32_16x16x4_f32(false, a1, false, b0, (short)0, acc10, false, false);
        acc11 = __builtin_amdgcn_wmma_f32_16x16x4_f32(false, a1, false, b1, (short)0, acc11, false, false);
    }

    // ---- epilogue: C 16x16 f32 layout: row = v + 8*(lane>>4), col = lane&15
    const int rowh = 8 * (lane >> 4);
    #pragma unroll
    for (int v = 0; v < 8; ++v) {
        int r0 = v + rowh;
        int r1 = 16 + v + rowh;
        atomicAdd(&proto[r0 * Cc + ch0], acc00[v]);
        atomicAdd(&proto[r0 * Cc + ch1], acc01[v]);
        if (r1 < Kc) {  // rows >= 19 are exactly zero, skip the atomics
            atomicAdd(&proto[r1 * Cc + ch0], acc10[v]);
            atomicAdd(&proto[r1 * Cc + ch1], acc11[v]);
        }
    }
}

// ---------------------------------------------------------------- finalize
__global__ __launch_bounds__(256)
void finalize_kernel(const float* __restrict__ proto, const unsigned* __restrict__ amask,
                     float* __restrict__ out) {
    __shared__ float kn[Kc * Cc];
    __shared__ float rnk[Kc];
    __shared__ float simm[Kc * Kc];
    __shared__ float sc[Kc];
    __shared__ float act[Kc];
    const int t = threadIdx.x;

    if (t < Kc) {
        float s = 0.0f;
        for (int c = 0; c < Cc; ++c) { float v = proto[t * Cc + c]; s += v * v; }
        rnk[t] = 1.0f / fmaxf(sqrtf(s), EPSF);
    }
    __syncthreads();
    for (int i = t; i < Kc * Cc; i += 256) kn[i] = proto[i] * rnk[i / Cc];
    __syncthreads();
    for (int ij = t; ij < Kc * Kc; ij += 256) {
        int i = ij / Kc, j = ij % Kc;
        float d = 0.0f;
        for (int c = 0; c < Cc; ++c) d += kn[i * Cc + c] * kn[j * Cc + c];
        simm[ij] = d / TAUF;
    }
    __syncthreads();
    const unsigned am = *amask;
    if (t < Kc) {
        float mx = -3.402823466e38f;
        for (int i = 0; i < Kc; ++i) mx = fmaxf(mx, simm[i * Kc + t]);
        float se = 0.0f;
        for (int i = 0; i < Kc; ++i) se += expf(simm[i * Kc + t] - mx);
        float lse = mx + logf(se);                 // logsumexp over axis 0 (column t)
        sc[t]  = simm[t * Kc + t] - lse;
        act[t] = ((am >> t) & 1u) ? 1.0f : 0.0f;
    }
    __syncthreads();
    if (t == 0) {
        float num = 0.0f, den = 0.0f;
        for (int k = 0; k < Kc; ++k) { num += act[k] * sc[k]; den += act[k]; }
        out[0] = -num / den;
    }
}

// ---------------------------------------------------------------- launch
extern "C" void kernel_launch(void* const* d_in, const int* in_sizes, int n_in,
                              void* d_out, int out_size, void* d_ws, size_t ws_size,
                              hipStream_t stream) {
    const float* feat = (const float*)d_in[0];
    const int*   gt   = (const int*)d_in[1];
    float*    proto = (float*)d_ws;                                    // 32*256 f32
    unsigned* amask = (unsigned*)((char*)d_ws + 32 * Cc * sizeof(float));

    init_ws_kernel<<<(32 * Cc + 255) / 256, 256, 0, stream>>>(proto, amask);
    proto_accum_kernel<<<NBLK, 256, 0, stream>>>(feat, gt, proto, amask);
    finalize_kernel<<<1, 256, 0, stream>>>(proto, amask, (float*)d_out);
}